// Q_MultiHeadedAttention_79602923864493
// MI455X (gfx1250) — compile-verified
//
#include <hip/hip_runtime.h>

typedef __attribute__((ext_vector_type(2))) float v2f;
typedef __attribute__((ext_vector_type(8))) float v8f;

struct cpx { float x, y; };

__device__ __forceinline__ cpx cmul(cpx a, cpx b) {
    cpx r; r.x = a.x*b.x - a.y*b.y; r.y = a.x*b.y + a.y*b.x; return r;
}
__device__ __forceinline__ cpx cadd(cpx a, cpx b) {
    cpx r; r.x = a.x + b.x; r.y = a.y + b.y; return r;
}
__device__ __forceinline__ cpx cscale(cpx a, float s) {
    cpx r; r.x = a.x*s; r.y = a.y*s; return r;
}
__device__ __forceinline__ float cabs2(cpx a) { return a.x*a.x + a.y*a.y; }

// Select M[r][c] for runtime r,c in {0,1} WITHOUT dynamic indexing
// (keeps the 2x2 matrices in VGPRs; lowers to v_cndmask, no scratch).
__device__ __forceinline__ cpx pick2x2(const cpx M[2][2], int r, int c) {
    cpx m00 = M[0][0], m01 = M[0][1], m10 = M[1][0], m11 = M[1][1];
    cpx top, bot, res;
    top.x = c ? m01.x : m00.x;  top.y = c ? m01.y : m00.y;
    bot.x = c ? m11.x : m10.x;  bot.y = c ? m11.y : m10.y;
    res.x = r ? bot.x : top.x;  res.y = r ? bot.y : top.y;
    return res;
}

// C = A * B (2x2 complex)
__device__ __forceinline__ void mm2(const cpx A[2][2], const cpx B[2][2], cpx C[2][2]) {
#pragma unroll
    for (int i = 0; i < 2; ++i)
#pragma unroll
        for (int j = 0; j < 2; ++j)
            C[i][j] = cadd(cmul(A[i][0], B[0][j]), cmul(A[i][1], B[1][j]));
}

// U = Rz(tz) @ Ry(ty) @ Rx(tx)  (circuit order rx,ry,rz).
// Angles are small (|t| < ~13): hardware v_sin/v_cos via __sincosf is accurate here
// and avoids the multi-KB libm argument-reduction path.
__device__ __forceinline__ void rot_u(float tx, float ty, float tz, cpx U[2][2]) {
    float cx, sx, cy, sy, cz, sz;
    __sincosf(tx * 0.5f, &sx, &cx);
    __sincosf(ty * 0.5f, &sy, &cy);
    __sincosf(tz * 0.5f, &sz, &cz);
    cpx Rx[2][2] = { { {cx,0.f}, {0.f,-sx} }, { {0.f,-sx}, {cx,0.f} } };
    cpx Ry[2][2] = { { {cy,0.f}, {-sy,0.f} }, { {sy,0.f}, {cy,0.f} } };
    cpx Rz[2][2] = { { {cz,-sz}, {0.f,0.f} }, { {0.f,0.f}, {cz,sz} } };
    cpx T[2][2];
    mm2(Ry, Rx, T);
    mm2(Rz, T, U);
}

__global__ __launch_bounds__(32)
void qattn_kernel(const float* __restrict__ q, const float* __restrict__ k,
                  const float* __restrict__ v, const float* __restrict__ wq,
                  const float* __restrict__ wk, const float* __restrict__ wv,
                  const float* __restrict__ lw, const float* __restrict__ lb,
                  float* __restrict__ out) {
    const int lane = threadIdx.x;   // one wave32

    __shared__ cpx  Uq[64], Uk[64], Uv[64];     // 8x8 complex unitaries
    __shared__ float qn[32], kn[32], vn[32];    // normalized rows [S=4][dq=8]
    __shared__ float dqg[32], dkg[32];          // |Uq q̂|^2, |Uk k̂|^2
    __shared__ cpx  yv[32];                     // Uv v̂  (complex, kept)
    __shared__ float sc[16];                    // scores [4][4]
    __shared__ float g[128];                    // g[i][j][m] = |u(φ)⊗3 · yv_j|^2
    __shared__ float xs[16][16];                // concat-head features, zero padded
    __shared__ float Bm[16][16];                // lin_w^T, zero padded

    // init WMMA operands
#pragma unroll
    for (int it = 0; it < 8; ++it) {
        int e = lane + it*32;
        int r = e >> 4, c = e & 15;
        xs[r][c] = 0.0f;
        Bm[r][c] = (c < 8) ? lw[c*16 + r] : 0.0f;   // Bm[in_dim][out_dim] = lin_w[out][in]
    }

    // normalize q,k,v rows (rho = outer(x,x)/||x||^2  ->  use x/||x||)
    if (lane < 12) {
        int fam = lane >> 2, i = lane & 3;
        const float* src = (fam == 0) ? q : (fam == 1) ? k : v;
        float*       dst = (fam == 0) ? qn : (fam == 1) ? kn : vn;
        float s = 0.f;
#pragma unroll
        for (int a = 0; a < 8; ++a) { float t = src[i*8 + a]; s += t*t; }
        float inv = rsqrtf(s);
#pragma unroll
        for (int a = 0; a < 8; ++a) dst[i*8 + a] = src[i*8 + a] * inv;
    }
    __syncthreads();

    const float WMUL = 0.6324555320336759f;  // sqrt(2)/sqrt(5)
    const float TWO_PI = 6.283185307179586f;

    for (int head = 0; head < 2; ++head) {
        // per-qubit 2x2 unitaries (redundant per lane, trivial)
        cpx uq[3][2][2], uk[3][2][2], uv[3][2][2];
#pragma unroll
        for (int qb = 0; qb < 3; ++qb) {
            int o = head*9 + 3*qb;
            rot_u(wq[o]*WMUL, wq[o+1]*WMUL, wq[o+2]*WMUL, uq[qb]);
            rot_u(wk[o]*WMUL, wk[o+1]*WMUL, wk[o+2]*WMUL, uk[qb]);
            rot_u(wv[o]*WMUL, wv[o+1]*WMUL, wv[o+2]*WMUL, uv[qb]);
        }
        // full 8x8 unitaries: U[m][a] = prod_q u_q[bit_q(m)][bit_q(a)], qubit0 = MSB
        // (2 entries per lane, selects instead of dynamic indexing -> no scratch)
#pragma unroll
        for (int it = 0; it < 2; ++it) {
            int e = lane + it*32;
            int m = e >> 3, a = e & 7;
            cpx pq = {1.f,0.f}, pk = {1.f,0.f}, pv = {1.f,0.f};
#pragma unroll
            for (int qb = 0; qb < 3; ++qb) {
                int mb = (m >> (2 - qb)) & 1, ab = (a >> (2 - qb)) & 1;
                pq = cmul(pq, pick2x2(uq[qb], mb, ab));
                pk = cmul(pk, pick2x2(uk[qb], mb, ab));
                pv = cmul(pv, pick2x2(uv[qb], mb, ab));
            }
            Uq[e] = pq; Uk[e] = pk; Uv[e] = pv;
        }
        __syncthreads();

        // y = U x̂ ; diag(rho)[m] = |y[m]|^2   (rank-1 density matrices)
        {
            int i = lane >> 3, m = lane & 7;
            cpx aq = {0.f,0.f}, ak = {0.f,0.f}, av = {0.f,0.f};
#pragma unroll
            for (int a = 0; a < 8; ++a) {
                aq = cadd(aq, cscale(Uq[m*8 + a], qn[i*8 + a]));
                ak = cadd(ak, cscale(Uk[m*8 + a], kn[i*8 + a]));
                av = cadd(av, cscale(Uv[m*8 + a], vn[i*8 + a]));
            }
            dqg[lane] = cabs2(aq);
            dkg[lane] = cabs2(ak);
            yv[lane]  = av;
        }
        __syncthreads();

        // scores[i][j] = (1/64) Σ_{hi,lo} (-1)^{pc(hi)+pc(lo)} dqg[i][hi] dkg[j][lo^hi]
        if (lane < 16) {
            int i = lane >> 2, j = lane & 3;
            float s = 0.f;
            for (int hi = 0; hi < 8; ++hi)
#pragma unroll
                for (int lo = 0; lo < 8; ++lo) {
                    float sgn = ((__popc(hi) + __popc(lo)) & 1) ? -1.f : 1.f;
                    s += sgn * dqg[i*8 + hi] * dkg[j*8 + (lo ^ hi)];
                }
            sc[lane] = s * (1.0f / 64.0f);
        }
        __syncthreads();

        // g[i][j][m] = | (u(φ)⊗u(φ)⊗u(φ)) · yv_j |^2[m],  u = Rz(φ)Ry(φ/2)Rx(φ/2)
        if (lane < 16) {
            int j = lane & 3;
            float phi = (sc[lane] - 0.5f) * TWO_PI;
            cpx u2[2][2];
            rot_u(phi*0.5f, phi*0.5f, phi, u2);
            cpx w[8];
#pragma unroll
            for (int m = 0; m < 8; ++m) w[m] = yv[j*8 + m];
#pragma unroll
            for (int qb = 0; qb < 3; ++qb) {
                int bit = 1 << (2 - qb);
#pragma unroll
                for (int m = 0; m < 8; ++m)
                    if (!(m & bit)) {
                        cpx w0 = w[m], w1 = w[m | bit];
                        w[m]       = cadd(cmul(u2[0][0], w0), cmul(u2[0][1], w1));
                        w[m | bit] = cadd(cmul(u2[1][0], w0), cmul(u2[1][1], w1));
                    }
            }
#pragma unroll
            for (int m = 0; m < 8; ++m) g[lane*8 + m] = cabs2(w[m]);
        }
        __syncthreads();

        // token output: XOR-convolution of the four g-vectors (exact partial trace)
        {
            int i = lane >> 3, m = lane & 7;
            const float* g0 = &g[(i*4 + 0)*8];
            const float* g1 = &g[(i*4 + 1)*8];
            const float* g2 = &g[(i*4 + 2)*8];
            const float* g3 = &g[(i*4 + 3)*8];
            float e8[8], c8[8];
#pragma unroll
            for (int t = 0; t < 8; ++t) {
                float s = 0.f;
#pragma unroll
                for (int a = 0; a < 8; ++a) s += g0[a] * g1[a ^ t];
                e8[t] = s;
            }
#pragma unroll
            for (int t = 0; t < 8; ++t) {
                float s = 0.f;
#pragma unroll
                for (int a = 0; a < 8; ++a) s += e8[a] * g2[a ^ t];
                c8[t] = s;
            }
            float s = 0.f;
#pragma unroll
            for (int t = 0; t < 8; ++t) s += c8[t] * g3[m ^ t];
            float z1 = (__popc(m) & 1) ? -1.f : 1.f;
            xs[i][head*8 + m] = z1 * s;
        }
        __syncthreads();
    }

    // Final projection: out[4,8] = xs[4,16] @ Bm[16,8(+pad)] + lb
    // Four chained V_WMMA_F32_16X16X4_F32 (K = 4 each -> K = 16), full f32.
    {
        int half = lane >> 4, r = lane & 15;
        v8f acc = {};
#pragma unroll
        for (int kk = 0; kk < 4; ++kk) {
            int K0 = 4*kk + 2*half;          // A frag: lane half selects K pair
            v2f a, b;
            a.x = xs[r][K0];  a.y = xs[r][K0 + 1];
            b.x = Bm[K0][r];  b.y = Bm[K0 + 1][r];
            acc = __builtin_amdgcn_wmma_f32_16x16x4_f32(
                false, a, false, b, (short)0, acc, false, false);
        }
        // D layout: lane L, VGPR r_v -> D[r_v + 8*(L/16)][L%16]
        if (half == 0 && r < 8) {
#pragma unroll
            for (int rv = 0; rv < 4; ++rv)
                out[rv*8 + r] = acc[rv] + lb[r];
        }
    }
}

extern "C" void kernel_launch(void* const* d_in, const int* in_sizes, int n_in,
                              void* d_out, int out_size, void* d_ws, size_t ws_size,
                              hipStream_t stream) {
    const float* q   = (const float*)d_in[0];
    const float* k   = (const float*)d_in[1];
    const float* v   = (const float*)d_in[2];
    const float* wq  = (const float*)d_in[3];
    const float* wk  = (const float*)d_in[4];
    const float* wv  = (const float*)d_in[5];
    const float* lw  = (const float*)d_in[6];
    const float* lb  = (const float*)d_in[7];
    float* out = (float*)d_out;
    qattn_kernel<<<dim3(1), dim3(32), 0, stream>>>(q, k, v, wq, wk, wv, lw, lb, out);
}